// Net_63986422776223
// MI455X (gfx1250) — compile-verified
//
#include <hip/hip_runtime.h>
#include <stdint.h>

// Problem constants (from reference: xyz [1, 64, 3, 262144] fp32, NPOINT=2).
#define BATCHES 64
#define CH 3
#define BLOCK 256
#define VEC 4
#define CHUNK (BLOCK * VEC)        // 1024 floats / channel / pipeline stage
#define BLOCKS_PER_BATCH 64        // 262144 / 64 = 4096 elems per block

// ---------------- CDNA5 feature probes (compile-always-clean) ----------------
#ifdef __has_builtin
#  if __has_builtin(__builtin_amdgcn_global_load_async_to_lds_b128)
#    define HAVE_ASYNC_LDS 1
#  else
#    define HAVE_ASYNC_LDS 0
#  endif
#  if __has_builtin(__builtin_amdgcn_s_wait_asynccnt)
#    define HAVE_WAIT_ASYNC 1
#  else
#    define HAVE_WAIT_ASYNC 0
#  endif
#else
#  define HAVE_ASYNC_LDS 0
#  define HAVE_WAIT_ASYNC 0
#endif

#if defined(__AMDGCN__) && !HAVE_ASYNC_LDS
#warning "CDNA5 probe: __builtin_amdgcn_global_load_async_to_lds_b128 NOT available (falling back to direct b128 loads)"
#endif
#if defined(__AMDGCN__) && !HAVE_WAIT_ASYNC
#warning "CDNA5 probe: __builtin_amdgcn_s_wait_asynccnt NOT available (inline asm fallback)"
#endif

#if HAVE_ASYNC_LDS
// Builtin signature (from clang diagnostic): param0 = v4i addrspace(1)*,
// param1 = v4i addrspace(3)*, then imm offset, imm cpol.
typedef int v4i __attribute__((ext_vector_type(4)));
typedef __attribute__((address_space(1))) v4i GV4;  // global int4*
typedef __attribute__((address_space(3))) v4i LV4;  // LDS int4*
#endif

template <int CNT>
__device__ __forceinline__ void wait_asynccnt() {
#if HAVE_WAIT_ASYNC
  __builtin_amdgcn_s_wait_asynccnt(CNT);
#else
  asm volatile("s_wait_asynccnt %0" ::"n"(CNT) : "memory");
#endif
}

// -------- packed argmax key: monotone float->u32 key, low 32 bits = ~index --
// larger value wins; on ties smaller index wins (matches jnp.argmax semantics)
__device__ __forceinline__ unsigned keyOf(float f) {
  unsigned u = __float_as_uint(f);
  return (u & 0x80000000u) ? ~u : (u | 0x80000000u);
}
__device__ __forceinline__ unsigned long long packKI(float f, unsigned idx) {
  return ((unsigned long long)keyOf(f) << 32) |
         (unsigned long long)(0xFFFFFFFFu - idx);
}
__device__ __forceinline__ unsigned unpackIdx(unsigned long long v) {
  return 0xFFFFFFFFu - (unsigned)(v & 0xFFFFFFFFull);
}
__device__ __forceinline__ unsigned long long maxu64(unsigned long long a,
                                                     unsigned long long b) {
  return a > b ? a : b;
}

__device__ __forceinline__ unsigned long long
blockReduceMax(unsigned long long v, unsigned long long* sm) {
  const int t = threadIdx.x;
  sm[t] = v;
  __syncthreads();
  for (int s = BLOCK >> 1; s > 0; s >>= 1) {
    if (t < s) sm[t] = maxu64(sm[t], sm[t + s]);
    __syncthreads();
  }
  return sm[0];
}

// ---------------------------------- kernels ---------------------------------

__global__ __launch_bounds__(128) void kInit(unsigned long long* ws) {
  if (threadIdx.x < 2 * BATCHES) ws[threadIdx.x] = 0ull;
}

// Pass 1: per-batch argmax over the y channel (channel 1). 67 MB streamed.
__global__ __launch_bounds__(BLOCK) void kArgmaxY(
    const float* __restrict__ xyz, unsigned long long* __restrict__ ws0,
    int N) {
  __shared__ unsigned long long sm[BLOCK];
  const int b = blockIdx.y;
  const float* __restrict__ y = xyz + ((size_t)b * CH + 1) * (size_t)N;
  const int per = N / gridDim.x;          // 4096 (N divisible by 64)
  const int start = blockIdx.x * per;
  unsigned long long best = 0;
  for (int i = start + threadIdx.x * VEC; i < start + per; i += BLOCK * VEC) {
    const float4 v = *(const float4*)(y + i);  // global_load_b128
    best = maxu64(best, packKI(v.x, i));
    best = maxu64(best, packKI(v.y, i + 1));
    best = maxu64(best, packKI(v.z, i + 2));
    best = maxu64(best, packKI(v.w, i + 3));
  }
  best = blockReduceMax(best, sm);
  if (threadIdx.x == 0) atomicMax(&ws0[b], best);  // global_atomic_max_u64
}

// Pass 2: per-batch argmax of squared distance to centroid0. 201 MB streamed.
// CDNA5 path: GLOBAL_LOAD_ASYNC_TO_LDS_B128 double-buffering, ASYNCcnt waits.
// Each lane consumes exactly the LDS bytes it loaded -> no block barrier
// needed between wait and use (per-wave ASYNCcnt suffices).
__global__ __launch_bounds__(BLOCK) void kArgmaxDist(
    const float* __restrict__ xyz, const unsigned long long* __restrict__ ws0,
    unsigned long long* __restrict__ ws1, int N) {
  __shared__ unsigned long long sm[BLOCK];
  const int b = blockIdx.y;
  const float* __restrict__ base = xyz + (size_t)b * CH * (size_t)N;
  const unsigned idx0 = unpackIdx(ws0[b]);
  const float cx = base[idx0];
  const float cy = base[(size_t)N + idx0];
  const float cz = base[2 * (size_t)N + idx0];
  const int per = N / gridDim.x;  // 4096
  const int start = blockIdx.x * per;
  const int t = threadIdx.x;
  unsigned long long best = 0;

#if HAVE_ASYNC_LDS
  __shared__ float buf[2][CH][CHUNK];  // 24 KB double buffer (of 320 KB/WGP)
  auto issue = [&](int it, int pbuf) {
    const int off = start + it * CHUNK + t * VEC;
#pragma unroll
    for (int c = 0; c < CH; ++c) {
      __builtin_amdgcn_global_load_async_to_lds_b128(
          (GV4*)(base + (size_t)c * (size_t)N + off),
          (LV4*)(&buf[pbuf][c][t * VEC]), /*offset=*/0, /*cpol=*/0);
    }
  };
  const int iters = per / CHUNK;  // 4
  int pb = 0;
  issue(0, 0);
  for (int it = 0; it < iters; ++it) {
    if (it + 1 < iters) {
      issue(it + 1, pb ^ 1);   // 3 more in flight
      wait_asynccnt<3>();      // oldest 3 (current buffer) retired
    } else {
      wait_asynccnt<0>();
    }
    const int off = start + it * CHUNK + t * VEC;
#pragma unroll
    for (int j = 0; j < VEC; ++j) {
      const float dx = buf[pb][0][t * VEC + j] - cx;
      const float dy = buf[pb][1][t * VEC + j] - cy;
      const float dz = buf[pb][2][t * VEC + j] - cz;
      const float d = fmaf(dx, dx, fmaf(dy, dy, dz * dz));
      best = maxu64(best, packKI(d, off + j));
    }
    pb ^= 1;
  }
#else
  for (int i = start + t * VEC; i < start + per; i += BLOCK * VEC) {
    __builtin_prefetch(base + i + BLOCK * VEC, 0, 0);  // global_prefetch_b8
    const float4 xv = *(const float4*)(base + i);
    const float4 yv = *(const float4*)(base + (size_t)N + i);
    const float4 zv = *(const float4*)(base + 2 * (size_t)N + i);
    const float xs[VEC] = {xv.x, xv.y, xv.z, xv.w};
    const float ys[VEC] = {yv.x, yv.y, yv.z, yv.w};
    const float zs[VEC] = {zv.x, zv.y, zv.z, zv.w};
#pragma unroll
    for (int j = 0; j < VEC; ++j) {
      const float dx = xs[j] - cx, dy = ys[j] - cy, dz = zs[j] - cz;
      const float d = fmaf(dx, dx, fmaf(dy, dy, dz * dz));
      best = maxu64(best, packKI(d, i + j));
    }
  }
#endif

  best = blockReduceMax(best, sm);
  if (t == 0) atomicMax(&ws1[b], best);
}

// Reference returns int64 [64, 2].
__global__ __launch_bounds__(64) void kFinal(
    const unsigned long long* __restrict__ ws, long long* __restrict__ out) {
  const int b = threadIdx.x;
  if (b < BATCHES) {
    out[2 * b] = (long long)unpackIdx(ws[b]);
    out[2 * b + 1] = (long long)unpackIdx(ws[BATCHES + b]);
  }
}

// -------------------------------- launcher ----------------------------------
extern "C" void kernel_launch(void* const* d_in, const int* in_sizes, int n_in,
                              void* d_out, int out_size, void* d_ws,
                              size_t ws_size, hipStream_t stream) {
  const float* xyz = (const float*)d_in[0];
  const int N = in_sizes[0] / (BATCHES * CH);  // 262144
  unsigned long long* ws = (unsigned long long*)d_ws;  // [2*BATCHES] u64

  kInit<<<1, 128, 0, stream>>>(ws);

  dim3 g(BLOCKS_PER_BATCH, BATCHES);
  kArgmaxY<<<g, BLOCK, 0, stream>>>(xyz, ws, N);
  kArgmaxDist<<<g, BLOCK, 0, stream>>>(xyz, ws, ws + BATCHES, N);

  kFinal<<<1, 64, 0, stream>>>(ws, (long long*)d_out);
}